// CLAMP_66726611910926
// MI455X (gfx1250) — compile-verified
//
#include <hip/hip_runtime.h>
#include <stdint.h>

// Sequential scan: one wave32 per algorithm row `a`.
//   lane owns columns n = lane + 32*j, j = 0..3   (N = 128)
// Output (A, N, T+1) staged in LDS tiles of TS time-steps and drained with
// CDNA5 global_store_async_from_lds_b128 (ASYNCcnt), double-buffered.
#define NCOL 128
#define TS   32            // time-steps per staged tile (T % TS == 0 assumed)
#define PAD  36            // LDS row pitch (floats): mult of 4 -> 16B-aligned b128 LDS side

#if __has_builtin(__builtin_amdgcn_tanhf)
__device__ __forceinline__ float fast_tanh(float x) {
  return __builtin_amdgcn_tanhf(x);            // v_tanh_f32 (gfx1250 trans op)
}
#else
__device__ __forceinline__ float fast_tanh(float x) {
  float r;
  asm volatile("v_tanh_f32 %0, %1" : "=v"(r) : "v"(x));
  return r;
}
#endif

__device__ __forceinline__ float bcast_lane(float v, int lane_uniform) {
  // wave-uniform lane index -> single v_readlane_b32 (no LDS round trip)
  return __int_as_float(__builtin_amdgcn_readlane(__float_as_int(v), lane_uniform));
}

__global__ void __launch_bounds__(32, 1)
SCAN_66726611910926_kernel(const int*   __restrict__ lx,
                           const float* __restrict__ tm,     // task_matrix (N x N)
                           const float* __restrict__ diff,   // (N)
                           const float* __restrict__ effA,   // (A)
                           const float* __restrict__ memA,   // (A)
                           const float* __restrict__ boostA, // (A)
                           float*       __restrict__ out,    // (A, N, T+1)
                           int T)
{
  __shared__ float stage[2 * NCOL * PAD];

  const int a    = blockIdx.x;
  const int lane = threadIdx.x;              // 0..31 (wave32)
  const int TOUT = T + 1;

  const float eff   = effA[a];
  const float mem   = memA[a];
  const float boost = boostA[a];

  // precompute 0.5/diff per owned column (once; off the hot loop)
  const float h0 = 0.5f / diff[lane +  0];
  const float h1 = 0.5f / diff[lane + 32];
  const float h2 = 0.5f / diff[lane + 64];
  const float h3 = 0.5f / diff[lane + 96];

  float* out_a = out + (size_t)a * NCOL * TOUT;
  const uint32_t lbase = (uint32_t)(uintptr_t)&stage[0];   // flat low 32b == LDS byte offset

  // tau = 0 row of tile 0 is s0 == 0
  stage[(uint32_t)(lane +  0) * PAD] = 0.0f;
  stage[(uint32_t)(lane + 32) * PAD] = 0.0f;
  stage[(uint32_t)(lane + 64) * PAD] = 0.0f;
  stage[(uint32_t)(lane + 96) * PAD] = 0.0f;

  // software pipeline (2 deep): carry row*eff and row*boost, not the raw row,
  // so only ONE fma sits between s_t arriving and r'.
  int ixn = (T > 1) ? lx[1] : 0;             // scalar (uniform) load
  const float* rp = tm + (size_t)lx[0] * NCOL + lane;
  float t0 = rp[0], t1 = rp[32], t2 = rp[64], t3 = rp[96];
  float ce0 = t0 * eff,   ce1 = t1 * eff,   ce2 = t2 * eff,   ce3 = t3 * eff;
  float cb0 = t0 * boost, cb1 = t1 * boost, cb2 = t2 * boost, cb3 = t3 * boost;
  rp = tm + (size_t)ixn * NCOL + lane;
  t0 = rp[0]; t1 = rp[32]; t2 = rp[64]; t3 = rp[96];
  float ne0 = t0 * eff,   ne1 = t1 * eff,   ne2 = t2 * eff,   ne3 = t3 * eff;
  float nb0 = t0 * boost, nb1 = t1 * boost, nb2 = t2 * boost, nb3 = t3 * boost;

  float r0 = 0.f, r1 = 0.f, r2 = 0.f, r3 = 0.f;
  float s0 = 0.f, s1 = 0.f, s2 = 0.f, s3 = 0.f;
  float s_t = 0.f;                           // s[a, lx[t]] from previous step

  for (int t = 0; t < T; ++t) {
    // ---- prefetch row for step t+2 (independent of the serial chain) ----
    const int ixf = (t + 2 < T) ? lx[t + 2] : 0;
    const float* fp = tm + (size_t)ixf * NCOL + lane;
    float f0 = fp[0], f1 = fp[32], f2 = fp[64], f3 = fp[96];
    float fe0 = f0 * eff,   fe1 = f1 * eff,   fe2 = f2 * eff,   fe3 = f3 * eff;
    float fb0 = f0 * boost, fb1 = f1 * boost, fb2 = f2 * boost, fb3 = f3 * boost;

    const int tau = t + 1;
    const int ts  = tau & (TS - 1);
    const uint32_t pb = ((uint32_t)(tau >> 5) & 1u) * (NCOL * PAD);  // buffer parity

    // ---- serial chain: fma(s_t) -> mul -> v_tanh -> select -> readlane ----
    // tmp = r*mem + c*eff does NOT depend on s_t (overlaps previous step's tail)
    r0 = fmaf(s_t, cb0, fmaf(r0, mem, ce0));  s0 = fast_tanh(r0 * h0);
    r1 = fmaf(s_t, cb1, fmaf(r1, mem, ce1));  s1 = fast_tanh(r1 * h1);
    r2 = fmaf(s_t, cb2, fmaf(r2, mem, ce2));  s2 = fast_tanh(r2 * h2);
    r3 = fmaf(s_t, cb3, fmaf(r3, mem, ce3));  s3 = fast_tanh(r3 * h3);

    // ---- stage to LDS (fire-and-forget, off the chain) ----
    stage[pb + (uint32_t)(lane +  0) * PAD + ts] = s0;
    stage[pb + (uint32_t)(lane + 32) * PAD + ts] = s1;
    stage[pb + (uint32_t)(lane + 64) * PAD + ts] = s2;
    stage[pb + (uint32_t)(lane + 96) * PAD + ts] = s3;

    // ---- tile full: drain via async LDS->global (512B per instruction) ----
    if (ts == TS - 1) {
      const int tau0 = tau - (TS - 1);
      const int cc = lane & 7;               // 4-float chunk within tile row
      const int nn = lane >> 3;              // first column handled by this lane
      const float* g = out_a + (size_t)nn * TOUT + tau0 + 4 * cc;
      uint32_t lv = lbase + (pb + (uint32_t)nn * PAD + 4u * (uint32_t)cc) * 4u;
      // LDS writes visible to async engine; previous tiles' asyncs drained
      asm volatile("s_wait_dscnt 0\n\ts_wait_asynccnt 0" ::: "memory");
#pragma unroll
      for (int i = 0; i < 32; ++i) {         // 32 instrs x 32 lanes x 16B = 16KB tile
        asm volatile("global_store_async_from_lds_b128 %0, %1, off"
                     :: "v"((uint64_t)(uintptr_t)g), "v"(lv)
                     : "memory");
        g  += 4 * (size_t)TOUT;              // +4 columns
        lv += 16u * PAD;
      }
    }

    // ---- wave-uniform broadcast of s[a, lx[t+1]] for the next step ----
    const int jsel = ixn >> 5;               // uniform: which of the 4 regs
    const int lsel = ixn & 31;               // uniform: which lane
    float va = (jsel & 2) ? ((jsel & 1) ? s3 : s2)
                          : ((jsel & 1) ? s1 : s0);
    s_t = bcast_lane(va, lsel);              // v_readlane_b32 (uniform index)

    // rotate pipeline registers
    ce0 = ne0; ce1 = ne1; ce2 = ne2; ce3 = ne3;
    cb0 = nb0; cb1 = nb1; cb2 = nb2; cb3 = nb3;
    ne0 = fe0; ne1 = fe1; ne2 = fe2; ne3 = fe3;
    nb0 = fb0; nb1 = fb1; nb2 = fb2; nb3 = fb3;
    ixn = ixf;
  }

  // tau == T lands at ts==0 of a fresh tile (T % TS == 0): store directly
  out_a[(size_t)(lane +  0) * TOUT + T] = s0;
  out_a[(size_t)(lane + 32) * TOUT + T] = s1;
  out_a[(size_t)(lane + 64) * TOUT + T] = s2;
  out_a[(size_t)(lane + 96) * TOUT + T] = s3;
}

extern "C" void kernel_launch(void* const* d_in, const int* in_sizes, int n_in,
                              void* d_out, int out_size, void* d_ws, size_t ws_size,
                              hipStream_t stream) {
  (void)n_in; (void)out_size; (void)d_ws; (void)ws_size;
  const int*   lx    = (const int*)  d_in[0];
  const float* tm    = (const float*)d_in[1];
  const float* diff  = (const float*)d_in[2];
  const float* eff   = (const float*)d_in[3];
  const float* mem   = (const float*)d_in[4];
  const float* boost = (const float*)d_in[5];
  float* out = (float*)d_out;

  const int T = in_sizes[0];      // 8192
  const int A = in_sizes[3];      // 32

  SCAN_66726611910926_kernel<<<A, 32, 0, stream>>>(lx, tm, diff, eff, mem, boost, out, T);
}